// ExcitationGeneration_48790828482879
// MI455X (gfx1250) — compile-verified
//
#include <hip/hip_runtime.h>
#include <math.h>

#define FP      80          // frame period (upsample factor)
#define NFRAMES 8192        // frames per row (fixed by reference)
#define CHUNK   8192        // samples per block
#define TPB     256         // threads per block (8 waves, wave32)
#define SPT     32          // samples per thread (CHUNK/TPB)
#define NEG_INF (-__builtin_inff())

typedef __attribute__((ext_vector_type(2))) float v2f;
typedef __attribute__((ext_vector_type(8))) float v8f;

struct FrameCtx { float a, step, qneg; int voiced, onset; };

// Per-frame interpolation context (boundary-fixed pitch, onset q flip value).
__device__ __forceinline__ FrameCtx frame_ctx(const float* __restrict__ pr, int k) {
  float pk   = pr[k];
  float pkm1 = (k > 0) ? pr[k - 1] : 0.0f;
  bool vk = pk > 0.0f, vkm1 = pkm1 > 0.0f;
  float pmk = (!vk && vkm1) ? pkm1 : pk;           // copy prev pitch into first unvoiced frame
  float pm_next;
  if (k + 1 < NFRAMES) {
    float pk1 = pr[k + 1];
    bool vk1 = pk1 > 0.0f;
    pm_next = (!vk1 && vk) ? pk : pk1;             // boundary fix on next frame too
  } else {
    pm_next = pmk;                                  // last frame replicated
  }
  FrameCtx f;
  f.voiced = vk ? 1 : 0;
  f.onset  = (vk && !vkm1) ? 1 : 0;
  f.a      = pmk;
  f.step   = (pm_next - pmk) * (1.0f / FP);
  float pu1 = pmk + f.step;                        // p_up at r==1
  f.qneg = (pu1 > 0.0f) ? (-1.0f / pu1) : 0.0f;    // q[0] = -q[1] at onset
  return f;
}

__device__ __forceinline__ void sample_qpu(const FrameCtx& f, int r, float& pu, float& q) {
  pu = f.voiced ? fmaf((float)r, f.step, f.a) : 0.0f;
  q  = (pu > 0.0f) ? (1.0f / pu) : 0.0f;
  if (r == 0 && f.onset) q = f.qneg;
}

// Sequential pass over SPT samples: Q = sum q, M = max over unvoiced of local
// inclusive prefix (-inf if none), Z = 0 if any voiced sample else -inf.
__device__ __forceinline__ void pass1(const float* __restrict__ pr, int g0,
                                      float& Q, float& M, float& Z) {
  float s = 0.0f, m = NEG_INF, z = NEG_INF;
  int curk = -1;
  FrameCtx f{};
  for (int j = 0; j < SPT; ++j) {
    int i = g0 + j;
    int k = i / FP;
    int r = i - k * FP;
    if (k != curk) { f = frame_ctx(pr, k); curk = k; }
    float pu, q;
    sample_qpu(f, r, pu, q);
    s += q;
    if (f.voiced) z = 0.0f;
    else          m = fmaxf(m, s);
  }
  Q = s; M = m; Z = z;
}

// ---------------- Kernel A: per-chunk (Q, M, Z) reduction ----------------
__global__ void __launch_bounds__(TPB)
exgen_chunk_reduce(const float* __restrict__ p, int cpr,
                   float* __restrict__ cQ, float* __restrict__ cM, float* __restrict__ cZ) {
  __shared__ float sQ[TPB], sM[TPB], sZ[TPB];
  int c   = blockIdx.x;
  int row = c / cpr;
  int cir = c - row * cpr;
  const float* pr = p + (size_t)row * NFRAMES;
  int tid = threadIdx.x;
  int g0  = cir * CHUNK + tid * SPT;
  __builtin_prefetch(pr + (g0 / FP), 0, 1);

  float Q, M, Z;
  pass1(pr, g0, Q, M, Z);
  sQ[tid] = Q; sM[tid] = M; sZ[tid] = Z;
  __syncthreads();
  // contiguous tree reduction with the (non-commutative) left∘right combine
  for (int st = 1; st < TPB; st <<= 1) {
    if ((tid & (2 * st - 1)) == 0) {
      float Q1 = sQ[tid],      M1 = sM[tid],      Z1 = sZ[tid];
      float Q2 = sQ[tid + st], M2 = sM[tid + st], Z2 = sZ[tid + st];
      sQ[tid] = Q1 + Q2;
      sM[tid] = fmaxf(M1, Q1 + M2);
      sZ[tid] = fmaxf(Z1, Z2);
    }
    __syncthreads();
  }
  if (tid == 0) { cQ[c] = sQ[0]; cM[c] = sM[0]; cZ[c] = sZ[0]; }
}

// ---------------- Kernel B: serial carry scan over chunks per row ----------------
__global__ void exgen_carry_scan(int rows, int cpr,
                                 const float* __restrict__ cQ, const float* __restrict__ cM,
                                 const float* __restrict__ cZ,
                                 float* __restrict__ cS, float* __restrict__ cB) {
  int r = blockIdx.x * blockDim.x + threadIdx.x;
  if (r >= rows) return;
  float S = 0.0f, Bb = NEG_INF;
  for (int c = 0; c < cpr; ++c) {
    int idx = r * cpr + c;
    cS[idx] = S;             // exclusive carry into chunk
    cB[idx] = Bb;
    float q = cQ[idx], m = cM[idx], z = cZ[idx];
    Bb = fmaxf(Bb, fmaxf(z, S + m));
    S += q;
  }
}

// ---------------- Kernel C: replay with carries, WMMA intra-block Q-scan, emit ----------------
__global__ void __launch_bounds__(TPB)
exgen_emit(const float* __restrict__ p, float* __restrict__ out, int cpr,
           const float* __restrict__ cS, const float* __restrict__ cB) {
  __shared__ float lds_e[CHUNK];     // staged output for coalesced stores
  __shared__ float lds_q[TPB];
  __shared__ float lds_incl[TPB];
  __shared__ float lds_seg[16];
  __shared__ float lds_v[TPB];

  int c   = blockIdx.x;
  int tid = threadIdx.x;
  int row = c / cpr;
  int cir = c - row * cpr;
  const float* pr = p + (size_t)row * NFRAMES;
  const int L = NFRAMES * FP;
  int g0 = cir * CHUNK + tid * SPT;
  __builtin_prefetch(pr + (g0 / FP), 0, 1);

  float Q, M, Z;
  pass1(pr, g0, Q, M, Z);
  lds_q[tid] = Q;
  __syncthreads();

  // --- WMMA matmul-scan: inclusive prefix of 256 Q values, done by wave 0. ---
  // X[k,n] = Q[16n + k]; C = L_tri(16x16) * X via 4 chained V_WMMA_F32_16X16X4_F32.
  if (tid < 32) {
    int l   = tid;
    int nlo = l & 15;           // A: row m; B: column n; C: column n
    int hi  = l >> 4;           // lane half selects K (A/B) / M+8 (C)
    v8f acc = {0, 0, 0, 0, 0, 0, 0, 0};
#pragma unroll
    for (int j = 0; j < 4; ++j) {
      int k0 = 4 * j + (hi ? 2 : 0);
      int k1 = 4 * j + (hi ? 3 : 1);
      v2f a, b;
      a.x = (k0 <= nlo) ? 1.0f : 0.0f;       // L[m, k0]
      a.y = (k1 <= nlo) ? 1.0f : 0.0f;       // L[m, k1]
      b.x = lds_q[(nlo << 4) + k0];          // X[k0, n]
      b.y = lds_q[(nlo << 4) + k1];          // X[k1, n]
      acc = __builtin_amdgcn_wmma_f32_16x16x4_f32(
          /*neg_a=*/false, a, /*neg_b=*/false, b,
          /*c_mod=*/(short)0, acc, /*reuse_a=*/false, /*reuse_b=*/false);
    }
#pragma unroll
    for (int j2 = 0; j2 < 8; ++j2) {
      int m = j2 + (hi ? 8 : 0);
      lds_incl[(nlo << 4) + m] = acc[j2];    // inclusive prefix within segment n
    }
  }
  __syncthreads();
  if (tid == 0) {                            // cross-segment offsets (16 adds)
    float run = 0.0f;
    for (int n = 0; n < 16; ++n) { lds_seg[n] = run; run += lds_incl[(n << 4) + 15]; }
  }
  __syncthreads();

  float sum_excl = lds_incl[tid] + lds_seg[tid >> 4] - Q;   // exclusive Q prefix in block
  float S_in = cS[c], B_in = cB[c];

  // --- exclusive max-scan of v_t = max(Z, S_abs + M) over 256 threads (LDS) ---
  float v = fmaxf(Z, S_in + sum_excl + M);
  lds_v[tid] = v;
  __syncthreads();
  for (int off = 1; off < TPB; off <<= 1) {
    float x = lds_v[tid];
    float y = (tid >= off) ? lds_v[tid - off] : NEG_INF;
    __syncthreads();
    lds_v[tid] = fmaxf(x, y);
    __syncthreads();
  }
  float vexcl = (tid > 0) ? lds_v[tid - 1] : NEG_INF;

  // --- pass 2: replay 32 samples + 1 lookahead, emit pulses ---
  float s = S_in + sum_excl;
  float b = fmaxf(B_in, vexcl);
  int curk = -1;
  FrameCtx f{};
  float sf_prev = 0.0f, pu_prev = 0.0f;
  for (int j = 0; j <= SPT; ++j) {
    int i = g0 + j;
    float sf, pu = 0.0f;
    if (i < L) {
      int k = i / FP;
      int r = i - k * FP;
      if (k != curk) { f = frame_ctx(pr, k); curk = k; }
      float q;
      sample_qpu(f, r, pu, q);
      s += q;
      b = fmaxf(b, f.voiced ? 0.0f : s);     // bias = cummax(mask ? 0 : s)
      sf = floorf(s - b);
    } else {
      sf = sf_prev;                           // last sample replicated -> diff 0
    }
    if (j > 0) {
      float e = ((sf - sf_prev) == 1.0f) ? sqrtf(pu_prev) : 0.0f;
      lds_e[tid * SPT + (j - 1)] = e;
    }
    sf_prev = sf;
    pu_prev = pu;
  }
  __syncthreads();

  // coalesced float4 stores of the whole chunk
  float4* o4 = (float4*)(out + (size_t)row * L + (size_t)cir * CHUNK);
  const float4* l4 = (const float4*)lds_e;
#pragma unroll
  for (int i2 = 0; i2 < CHUNK / 4 / TPB; ++i2) {
    o4[i2 * TPB + tid] = l4[i2 * TPB + tid];
  }
}

extern "C" void kernel_launch(void* const* d_in, const int* in_sizes, int n_in,
                              void* d_out, int out_size, void* d_ws, size_t ws_size,
                              hipStream_t stream) {
  const float* p = (const float*)d_in[0];
  float* out = (float*)d_out;

  int total  = in_sizes[0];              // B * N
  int rows   = total / NFRAMES;          // 32
  int L      = NFRAMES * FP;             // 655360 samples per row
  int cpr    = L / CHUNK;                // 80 chunks per row
  int nchunk = rows * cpr;               // 2560

  float* ws = (float*)d_ws;              // ~50 KB workspace
  float* cQ = ws;
  float* cM = cQ + nchunk;
  float* cZ = cM + nchunk;
  float* cS = cZ + nchunk;
  float* cB = cS + nchunk;

  exgen_chunk_reduce<<<nchunk, TPB, 0, stream>>>(p, cpr, cQ, cM, cZ);
  exgen_carry_scan<<<1, 64, 0, stream>>>(rows, cpr, cQ, cM, cZ, cS, cB);
  exgen_emit<<<nchunk, TPB, 0, stream>>>(p, out, cpr, cS, cB);
}